// ContentEncoder_76948634075499
// MI455X (gfx1250) — compile-verified
//
#include <hip/hip_runtime.h>
#include <hip/hip_bf16.h>
#include <math.h>

typedef float v2f __attribute__((ext_vector_type(2)));
typedef float v8f __attribute__((ext_vector_type(8)));

#define GEMM_WAVES 4   // 128 threads; block tile = 16(M) x 64(N)
#define KC 32          // K chunk staged through LDS (8 WMMA steps)
#define LDA 36         // LDS row stride (floats): 16B-aligned, bank-conflict-free

// ---------------------------------------------------------------------------
// Weight packing: w[Mreal, Kd] row-major -> ap[Mp, Kdp] zero-padded.
// ---------------------------------------------------------------------------
__global__ void pack_weights_kernel(const float* __restrict__ w,
                                    float* __restrict__ ap,
                                    int Mreal, int Kd, int Mp, int Kdp) {
  long long total = (long long)Mp * Kdp;
  long long idx = (long long)blockIdx.x * blockDim.x + threadIdx.x;
  if (idx >= total) return;
  int m = (int)(idx / Kdp);
  int k = (int)(idx - (long long)m * Kdp);
  ap[idx] = (m < Mreal && k < Kd) ? w[(long long)m * Kd + k] : 0.f;
}

// ---------------------------------------------------------------------------
// Plain zero-pad im2col -> interleaved B layout [Kdp/4][Nc][4].
// Linear index IS the interleaved store address (coalesced writes):
//   idx = ((r>>2)*Nc + c)*4 + (r&3)
// ---------------------------------------------------------------------------
__global__ void im2col_plain_kernel(const float* __restrict__ x,
                                    float* __restrict__ col,
                                    int N, int Cin, int H, int W,
                                    int kh, int kw, int stride, int pad,
                                    int Ho, int Wo, int Kdp, long long total) {
  long long idx = (long long)blockIdx.x * blockDim.x + threadIdx.x;
  if (idx >= total) return;
  int HoWo = Ho * Wo;
  long long Nc = (long long)N * HoWo;
  int sub = (int)(idx & 3);
  long long q = idx >> 2;
  int rg = (int)(q / Nc);
  long long c = q - (long long)rg * Nc;
  int r = rg * 4 + sub;
  int K = kh * kw, CinK = Cin * K;

  float v = 0.f;
  if (r < CinK) {
    int n   = (int)(c / HoWo);
    int pix = (int)(c - (long long)n * HoWo);
    int ho = pix / Wo, wo = pix - ho * Wo;
    int ci = r / K, k = r - ci * K;
    int ky = k / kw, kx = k - ky * kw;
    int iy = ho * stride - pad + ky;
    int ix = wo * stride - pad + kx;
    if (iy >= 0 && iy < H && ix >= 0 && ix < W)
      v = x[((long long)(n * Cin + ci) * H + iy) * W + ix];
  }
  col[idx] = v;
}

// ---------------------------------------------------------------------------
// Deformable-conv bilinear sampling -> interleaved B layout (same addressing).
// off layout: [N, 3K, Ho, Wo]; rows [0,K)=oy, [K,2K)=ox, [2K,3K)=mask(raw)
// ---------------------------------------------------------------------------
__global__ void deform_im2col_kernel(const float* __restrict__ x,
                                     const float* __restrict__ off,
                                     float* __restrict__ col,
                                     int N, int Cin, int H, int W,
                                     int kh, int kw, int stride, int pad,
                                     int Ho, int Wo, int Kdp, long long total) {
  long long idx = (long long)blockIdx.x * blockDim.x + threadIdx.x;
  if (idx >= total) return;
  int HoWo = Ho * Wo;
  long long Nc = (long long)N * HoWo;
  int sub = (int)(idx & 3);
  long long q = idx >> 2;
  int rg = (int)(q / Nc);
  long long c = q - (long long)rg * Nc;
  int r = rg * 4 + sub;
  int K = kh * kw, CinK = Cin * K;

  float v = 0.f;
  if (r < CinK) {
    int n   = (int)(c / HoWo);
    int pix = (int)(c - (long long)n * HoWo);
    int ho = pix / Wo, wo = pix - ho * Wo;
    int ci = r / K, k = r - ci * K;
    int ky = k / kw, kx = k - ky * kw;

    const float* ob = off + (long long)n * (3 * K) * HoWo;
    float oy = ob[(long long)k * HoWo + pix];
    float ox = ob[(long long)(K + k) * HoWo + pix];
    float mr = ob[(long long)(2 * K + k) * HoWo + pix];
    float m  = 1.0f / (1.0f + expf(-mr));

    float py = (float)(ho * stride - pad + ky) + oy;
    float px = (float)(wo * stride - pad + kx) + ox;
    float fy0 = floorf(py), fx0 = floorf(px);
    float wy = py - fy0, wx = px - fx0;
    int y0 = (int)fy0, x0 = (int)fx0;

    const float* xp = x + ((long long)(n * Cin + ci)) * H * W;
    float v00 = (y0   >= 0 && y0   < H && x0   >= 0 && x0   < W) ? xp[y0 * W + x0]           : 0.f;
    float v01 = (y0   >= 0 && y0   < H && x0+1 >= 0 && x0+1 < W) ? xp[y0 * W + x0 + 1]       : 0.f;
    float v10 = (y0+1 >= 0 && y0+1 < H && x0   >= 0 && x0   < W) ? xp[(y0 + 1) * W + x0]     : 0.f;
    float v11 = (y0+1 >= 0 && y0+1 < H && x0+1 >= 0 && x0+1 < W) ? xp[(y0 + 1) * W + x0 + 1] : 0.f;

    float s = v00 * (1.f - wy) * (1.f - wx) + v01 * (1.f - wy) * wx +
              v10 * wy * (1.f - wx)         + v11 * wy * wx;
    v = s * m;
  }
  col[idx] = v;
}

// ---------------------------------------------------------------------------
// Reflect-pad-1 3x3 im2col (res blocks) -> interleaved B layout.
// ---------------------------------------------------------------------------
__global__ void im2col3_reflect_kernel(const float* __restrict__ x,
                                       float* __restrict__ col,
                                       int N, int C, int H, int W) {
  int HW = H * W;
  long long Nc = (long long)N * HW;
  long long total = (long long)C * 9 * Nc;
  long long idx = (long long)blockIdx.x * blockDim.x + threadIdx.x;
  if (idx >= total) return;
  int sub = (int)(idx & 3);
  long long q = idx >> 2;
  int rg = (int)(q / Nc);
  long long c = q - (long long)rg * Nc;
  int r = rg * 4 + sub;

  int n   = (int)(c / HW);
  int pix = (int)(c - (long long)n * HW);
  int ho = pix / W, wo = pix - ho * W;
  int ci = r / 9, k = r - ci * 9;
  int ky = k / 3, kx = k - ky * 3;
  int yy = ho + ky - 1;
  yy = yy < 0 ? -yy : (yy >= H ? 2 * H - 2 - yy : yy);
  int xx = wo + kx - 1;
  xx = xx < 0 ? -xx : (xx >= W ? 2 * W - 2 - xx : xx);
  col[idx] = x[((long long)(n * C + ci) * H + yy) * W + xx];
}

// ---------------------------------------------------------------------------
// fp32 WMMA GEMM, software-pipelined:
//   - A panel double-buffered in LDS (shared by the 4 waves)
//   - B chunk double-buffered in registers (8 independent b64 loads per chunk,
//     common base per chunk -> no serialized pointer-increment chain)
//   - next-chunk loads issued before current-chunk WMMAs -> loads overlap math
//   - one barrier per K chunk; EXEC all-ones at every WMMA
// Output written NCHW with bias; M padding handled by store-side guards only.
// ---------------------------------------------------------------------------
__global__ __launch_bounds__(32 * GEMM_WAVES)
void wmma_gemm_nchw_kernel(const float* __restrict__ A,
                           const float* __restrict__ Bm,
                           const float* __restrict__ bias,
                           float* __restrict__ out,
                           int Mreal, int Kdp, long long Nc, int HoWo) {
  __shared__ float lds_a[2][16 * LDA];
  const int tid  = threadIdx.x;
  const int lane = tid & 31;
  const int wave = tid >> 5;
  const int half = lane >> 4;   // K-pair selector within a 4-wide K step
  const int l    = lane & 15;   // row (A) / col (B) within 16-wide tile
  const int m0 = blockIdx.y * 16;
  const long long n0 = ((long long)blockIdx.x * GEMM_WAVES + wave) * 16;

  v8f acc = {};

  // Cooperative A load mapping: 128 threads x float4 = 16 rows x 32 K per chunk
  const int ar = tid >> 3;          // 0..15
  const int ac = (tid & 7) * 4;     // 0,4,...,28
  const float* ap = A + (long long)(m0 + ar) * Kdp + ac;
  const long long bstep = 4 * Nc;   // floats between consecutive K-steps
  const float* bbase = Bm + ((n0 + l) << 2) + 2 * half;

  // ---- prologue: chunk 0 -> LDS buf 0, B chunk 0 -> registers ----
  v2f breg[KC / 4];
  {
    float4 a4 = *(const float4*)(ap);
    *(float4*)&lds_a[0][ar * LDA + ac] = a4;
#pragma unroll
    for (int s = 0; s < KC / 4; ++s)
      breg[s] = *(const v2f*)(bbase + (long long)s * bstep);
    __syncthreads();
  }

  int parity = 0;
  for (int kk = 0; kk < Kdp; kk += KC) {
    // Issue ALL next-chunk loads first (clamped dummy refetch on last chunk:
    // uniform scalar select, in-bounds, never consumed).
    const int nk = (kk + KC < Kdp) ? (kk + KC) : 0;
    float4 a4n = *(const float4*)(ap + nk);
    const float* bq = bbase + (long long)(nk >> 2) * bstep;
    v2f bn[KC / 4];
#pragma unroll
    for (int s = 0; s < KC / 4; ++s)
      bn[s] = *(const v2f*)(bq + (long long)s * bstep);
    __builtin_prefetch(bq + (long long)(KC / 2) * bstep, 0, 1);  // ~2 chunks ahead

    // Compute current chunk from LDS + registers (overlaps the loads above).
    const float* lsrc = &lds_a[parity][l * LDA + 2 * half];
#pragma unroll
    for (int s = 0; s < KC / 4; ++s) {
      v2f a = *(const v2f*)(lsrc + 4 * s);
      acc = __builtin_amdgcn_wmma_f32_16x16x4_f32(
          false, a, false, breg[s], (short)0, acc, false, false);
    }

    // Stage next chunk; single barrier per iteration (double-buffered LDS).
    parity ^= 1;
    *(float4*)&lds_a[parity][ar * LDA + ac] = a4n;
#pragma unroll
    for (int s = 0; s < KC / 4; ++s) breg[s] = bn[s];
    __syncthreads();
  }

  long long ncol = n0 + l;
  int nimg = (int)(ncol / HoWo);   // tiles never straddle images (HoWo % 64 == 0)
  int pix  = (int)(ncol - (long long)nimg * HoWo);
  float* obase = out + (long long)nimg * Mreal * HoWo + pix;
#pragma unroll
  for (int r = 0; r < 8; ++r) {
    int m = m0 + r + half * 8;
    if (m < Mreal) obase[(long long)m * HoWo] = acc[r] + bias[m];
  }
}

// ---------------------------------------------------------------------------
// Instance norm over HW per (n,c), optional ReLU / residual add.
// ---------------------------------------------------------------------------
__global__ void instnorm_kernel(const float* __restrict__ in,
                                float* __restrict__ out,
                                const float* __restrict__ resid,
                                int HW, int do_relu) {
  const float* p = in + (long long)blockIdx.x * HW;
  float s = 0.f, q = 0.f;
  for (int i = threadIdx.x; i < HW; i += blockDim.x) {
    float v = p[i];
    s += v; q += v * v;
  }
  __shared__ float sh_s[256];
  __shared__ float sh_q[256];
  sh_s[threadIdx.x] = s;
  sh_q[threadIdx.x] = q;
  __syncthreads();
  for (int o = blockDim.x >> 1; o > 0; o >>= 1) {
    if ((int)threadIdx.x < o) {
      sh_s[threadIdx.x] += sh_s[threadIdx.x + o];
      sh_q[threadIdx.x] += sh_q[threadIdx.x + o];
    }
    __syncthreads();
  }
  float mean = sh_s[0] / (float)HW;
  float var  = sh_q[0] / (float)HW - mean * mean;
  float inv  = rsqrtf(var + 1e-5f);
  float* w = out + (long long)blockIdx.x * HW;
  const float* rz = resid ? resid + (long long)blockIdx.x * HW : nullptr;
  for (int i = threadIdx.x; i < HW; i += blockDim.x) {
    float v = (p[i] - mean) * inv;
    if (rz) v += rz[i];
    if (do_relu) v = fmaxf(v, 0.f);
    w[i] = v;
  }
}

// ---------------------------------------------------------------------------
// Host-side orchestration
// ---------------------------------------------------------------------------
static inline long long cdiv(long long a, long long b) { return (a + b - 1) / b; }

extern "C" void kernel_launch(void* const* d_in, const int* in_sizes, int n_in,
                              void* d_out, int out_size, void* d_ws, size_t ws_size,
                              hipStream_t stream) {
  const float* x      = (const float*)d_in[0];
  const float* off1_w = (const float*)d_in[1];
  const float* off1_b = (const float*)d_in[2];
  const float* dcn1_w = (const float*)d_in[3];
  const float* dcn1_b = (const float*)d_in[4];
  const float* off2_w = (const float*)d_in[5];
  const float* off2_b = (const float*)d_in[6];
  const float* dcn2_w = (const float*)d_in[7];
  const float* dcn2_b = (const float*)d_in[8];
  const float* off3_w = (const float*)d_in[9];
  const float* off3_b = (const float*)d_in[10];
  const float* dcn3_w = (const float*)d_in[11];
  const float* dcn3_b = (const float*)d_in[12];
  const float* rb1_w1 = (const float*)d_in[13];
  const float* rb1_b1 = (const float*)d_in[14];
  const float* rb1_w2 = (const float*)d_in[15];
  const float* rb1_b2 = (const float*)d_in[16];
  const float* rb2_w1 = (const float*)d_in[17];
  const float* rb2_b1 = (const float*)d_in[18];
  const float* rb2_w2 = (const float*)d_in[19];
  const float* rb2_b2 = (const float*)d_in[20];

  float* outp  = (float*)d_out;
  float* hfin  = outp;                 // (2,256,64,64)   = 2,097,152
  float* skip1 = outp + 2097152;       // (2,64,256,256)  = 8,388,608
  float* skip2 = skip1 + 8388608;      // (2,128,128,128) = 4,194,304

  float* ws      = (float*)d_ws;
  float* col     = ws;                 // up to 1024*32768 = 33,554,432
  float* off     = col + 33554432;     // up to 2*147*65536 = 19,267,584
  float* t0      = off + 19267584;     // GEMM out pre-norm, up to 8,388,608
  float* t1      = t0 + 8388608;       // 2,097,152
  float* t2      = t1 + 2097152;       // 2,097,152
  float* ap_off1 = t2 + 2097152;       // 160*160 = 25,600
  float* ap_dcn1 = ap_off1 + 25600;    // 64*160  = 10,240

  const int TB = 256;

  // ========== Stage 1: mdcp(x), k=7, s=1, p=3; Kd 147 -> 160 padded ==========
  {
    const int N=2, Cin=3, H=256, W=256, Ho=256, Wo=256;
    const int Kdp = 160;                       // 147 padded to 32-multiple
    const long long Nc = (long long)N * Ho * Wo;        // 131072
    const long long tot = (long long)Kdp * Nc;

    pack_weights_kernel<<<(unsigned)cdiv(160LL*160, TB), TB, 0, stream>>>(
        off1_w, ap_off1, 147, 147, 160, 160);
    pack_weights_kernel<<<(unsigned)cdiv(64LL*160, TB), TB, 0, stream>>>(
        dcn1_w, ap_dcn1, 64, 147, 64, 160);

    // Offset/mask conv as GEMM (M padded 147->160, store-guarded)
    im2col_plain_kernel<<<(unsigned)cdiv(tot, TB), TB, 0, stream>>>(
        x, col, N, Cin, H, W, 7, 7, 1, 3, Ho, Wo, Kdp, tot);
    {
      dim3 g((unsigned)(Nc / 64), 160 / 16);
      wmma_gemm_nchw_kernel<<<g, 32 * GEMM_WAVES, 0, stream>>>(
          ap_off1, col, off1_b, off, 147, Kdp, Nc, Ho * Wo);
    }

    // Deformable sampling + GEMM
    deform_im2col_kernel<<<(unsigned)cdiv(tot, TB), TB, 0, stream>>>(
        x, off, col, N, Cin, H, W, 7, 7, 1, 3, Ho, Wo, Kdp, tot);
    {
      dim3 g((unsigned)(Nc / 64), 64 / 16);
      wmma_gemm_nchw_kernel<<<g, 32 * GEMM_WAVES, 0, stream>>>(
          ap_dcn1, col, dcn1_b, t0, 64, Kdp, Nc, Ho * Wo);
    }
    instnorm_kernel<<<N * 64, TB, 0, stream>>>(t0, skip1, nullptr, Ho * Wo, 1);
  }

  // ========== Stage 2: mdcp(skip1), k=4, s=2, p=1; Kd = 1024 ==========
  {
    const int N=2, Cin=64, H=256, W=256, Ho=128, Wo=128;
    const int Kdp = 1024;
    const long long Nc = (long long)N * Ho * Wo;        // 32768
    const long long tot = (long long)Kdp * Nc;

    im2col_plain_kernel<<<(unsigned)cdiv(tot, TB), TB, 0, stream>>>(
        skip1, col, N, Cin, H, W, 4, 4, 2, 1, Ho, Wo, Kdp, tot);
    {
      dim3 g((unsigned)(Nc / 64), 48 / 16);
      wmma_gemm_nchw_kernel<<<g, 32 * GEMM_WAVES, 0, stream>>>(
          off2_w, col, off2_b, off, 48, Kdp, Nc, Ho * Wo);
    }
    deform_im2col_kernel<<<(unsigned)cdiv(tot, TB), TB, 0, stream>>>(
        skip1, off, col, N, Cin, H, W, 4, 4, 2, 1, Ho, Wo, Kdp, tot);
    {
      dim3 g((unsigned)(Nc / 64), 128 / 16);
      wmma_gemm_nchw_kernel<<<g, 32 * GEMM_WAVES, 0, stream>>>(
          dcn2_w, col, dcn2_b, t0, 128, Kdp, Nc, Ho * Wo);
    }
    instnorm_kernel<<<N * 128, TB, 0, stream>>>(t0, skip2, nullptr, Ho * Wo, 1);
  }

  // ========== Stage 3: mdcp(skip2), k=4, s=2, p=1; Kd = 2048 ==========
  {
    const int N=2, Cin=128, H=128, W=128, Ho=64, Wo=64;
    const int Kdp = 2048;
    const long long Nc = (long long)N * Ho * Wo;        // 8192
    const long long tot = (long long)Kdp * Nc;

    im2col_plain_kernel<<<(unsigned)cdiv(tot, TB), TB, 0, stream>>>(
        skip2, col, N, Cin, H, W, 4, 4, 2, 1, Ho, Wo, Kdp, tot);
    {
      dim3 g((unsigned)(Nc / 64), 48 / 16);
      wmma_gemm_nchw_kernel<<<g, 32 * GEMM_WAVES, 0, stream>>>(
          off3_w, col, off3_b, off, 48, Kdp, Nc, Ho * Wo);
    }
    deform_im2col_kernel<<<(unsigned)cdiv(tot, TB), TB, 0, stream>>>(
        skip2, off, col, N, Cin, H, W, 4, 4, 2, 1, Ho, Wo, Kdp, tot);
    {
      dim3 g((unsigned)(Nc / 64), 256 / 16);
      wmma_gemm_nchw_kernel<<<g, 32 * GEMM_WAVES, 0, stream>>>(
          dcn3_w, col, dcn3_b, t0, 256, Kdp, Nc, Ho * Wo);
    }
    instnorm_kernel<<<N * 256, TB, 0, stream>>>(t0, t1, nullptr, Ho * Wo, 1); // h3 -> t1
  }

  // ========== Res blocks (2, 256, 64, 64); Kd = 2304 ==========
  {
    const int N=2, C=256, H=64, W=64, HW=H*W;
    const int Kdp = 2304;
    const long long Nc = (long long)N * HW;             // 8192
    const long long tot = (long long)Kdp * Nc;
    dim3 gg((unsigned)(Nc / 64), C / 16);

    // --- rb1: in = t1, out = t2 ---
    im2col3_reflect_kernel<<<(unsigned)cdiv(tot, TB), TB, 0, stream>>>(t1, col, N, C, H, W);
    wmma_gemm_nchw_kernel<<<gg, 32 * GEMM_WAVES, 0, stream>>>(
        rb1_w1, col, rb1_b1, t0, C, Kdp, Nc, HW);
    instnorm_kernel<<<N * C, TB, 0, stream>>>(t0, t2, nullptr, HW, 1);

    im2col3_reflect_kernel<<<(unsigned)cdiv(tot, TB), TB, 0, stream>>>(t2, col, N, C, H, W);
    wmma_gemm_nchw_kernel<<<gg, 32 * GEMM_WAVES, 0, stream>>>(
        rb1_w2, col, rb1_b2, t0, C, Kdp, Nc, HW);
    instnorm_kernel<<<N * C, TB, 0, stream>>>(t0, t2, /*resid=*/t1, HW, 0);

    // --- rb2: in = t2, out = hfin ---
    im2col3_reflect_kernel<<<(unsigned)cdiv(tot, TB), TB, 0, stream>>>(t2, col, N, C, H, W);
    wmma_gemm_nchw_kernel<<<gg, 32 * GEMM_WAVES, 0, stream>>>(
        rb2_w1, col, rb2_b1, t0, C, Kdp, Nc, HW);
    instnorm_kernel<<<N * C, TB, 0, stream>>>(t0, t1, nullptr, HW, 1);

    im2col3_reflect_kernel<<<(unsigned)cdiv(tot, TB), TB, 0, stream>>>(t1, col, N, C, H, W);
    wmma_gemm_nchw_kernel<<<gg, 32 * GEMM_WAVES, 0, stream>>>(
        rb2_w2, col, rb2_b2, t0, C, Kdp, Nc, HW);
    instnorm_kernel<<<N * C, TB, 0, stream>>>(t0, hfin, /*resid=*/t2, HW, 0);
  }

  (void)in_sizes; (void)n_in; (void)out_size; (void)ws_size;
}